// TopKPool_20847771254945
// MI455X (gfx1250) — compile-verified
//
#include <hip/hip_runtime.h>

// Problem constants (match reference)
#define Bn 8
#define Cn 128
#define Tn 64
#define Vn 1024
#define Kn 512   // floor(0.5 * 1024)

typedef __attribute__((ext_vector_type(2))) float v2f;
typedef __attribute__((ext_vector_type(8))) float v8f;

// ---------------------------------------------------------------------------
// Kernel 1: scores[b,v] = sigmoid( (1/T) * sum_{c,t} X[b,c,t,v]*W[c] + bias )
// One wave computes 16 vertices via V_WMMA_F32_16X16X4_F32.
// A-matrix: all 64 elements of the 16x4 tile = W[c] (uniform per chunk, since
// a chunk of 4 consecutive k = c*T+t never crosses a channel boundary).
// B-matrix: the 4x16 tile of X rows k0..k0+3 for 16 consecutive vertices.
// Because A is uniform, any internal K-permutation of B is harmless; the
// N<->lane mapping (lanes 0-15 = N=0..15) is the documented layout.
// ---------------------------------------------------------------------------
__global__ __launch_bounds__(256) void score_wmma_kernel(
    const float* __restrict__ X, const float* __restrict__ W,
    const float* __restrict__ bias, float* __restrict__ scores)
{
    const int wave = threadIdx.x >> 5;
    const int lane = threadIdx.x & 31;
    const int half = lane >> 4;        // 0: K rows {t0, t0+1}, 1: {t0+2, t0+3}
    const int l15  = lane & 15;        // vertex within the 16-wide tile
    const int b    = blockIdx.x >> 3;  // 8 vertex-chunks of 128 per batch
    const int vch  = blockIdx.x & 7;
    const int vbase = vch * 128 + wave * 16;

    const float* xb = X + (size_t)b * Cn * Tn * Vn;

    v8f acc = {0.f, 0.f, 0.f, 0.f, 0.f, 0.f, 0.f, 0.f};
    for (int c = 0; c < Cn; ++c) {
        const float wc = W[c];
        v2f a; a.x = wc; a.y = wc;     // entire 16x4 A tile == wc
        const float* xc = xb + (size_t)c * Tn * Vn
                             + (size_t)half * 2 * Vn + vbase + l15;
        #pragma unroll 4
        for (int t0 = 0; t0 < Tn; t0 += 4) {
            v2f bm;
            bm.x = xc[(size_t)t0 * Vn];        // K row (t0 + 2*half)
            bm.y = xc[(size_t)t0 * Vn + Vn];   // K row (t0 + 2*half + 1)
            acc = __builtin_amdgcn_wmma_f32_16x16x4_f32(
                      false, a, false, bm, (short)0, acc, false, false);
        }
    }
    // D layout: VGPR0, lanes 0-15 -> (M=0, N=lane); all rows identical.
    if (lane < 16) {
        float s = acc[0] * (1.0f / (float)Tn) + bias[0];
        scores[b * Vn + vbase + lane] = 1.0f / (1.0f + expf(-s));
    }
}

// ---------------------------------------------------------------------------
// Kernel 2: per-batch top-K by ranking (strict total order, matches lax.top_k
// tie-breaking: higher score first, lower index wins ties).
//   values_ws[b, rank]  = score          (descending order automatically)
//   idx_ws  [b, pos]    = v              (ascending vertex ids, pos = prefix
//                                         count of selected ids < v)
// Also writes the float-cast idx output.
// ---------------------------------------------------------------------------
__global__ __launch_bounds__(1024) void topk_kernel(
    const float* __restrict__ scores, float* __restrict__ values_ws,
    int* __restrict__ idx_ws, float* __restrict__ out_idx)
{
    __shared__ float s[Vn];
    __shared__ int   sel[Vn];
    const int b = blockIdx.x;
    const int v = threadIdx.x;

    s[v] = scores[b * Vn + v];
    __syncthreads();

    const float sv = s[v];
    int r = 0;
    for (int u = 0; u < Vn; ++u) {
        const float su = s[u];
        r += (su > sv) || (su == sv && u < v);
    }
    const int isSel = (r < Kn) ? 1 : 0;
    sel[v] = isSel;
    if (isSel) values_ws[b * Kn + r] = sv;   // ranks of selected = {0..K-1}
    __syncthreads();

    if (isSel) {
        int pos = 0;
        for (int u = 0; u < v; ++u) pos += sel[u];
        idx_ws [b * Kn + pos] = v;
        out_idx[b * Kn + pos] = (float)v;
    }
}

// ---------------------------------------------------------------------------
// Kernel 3: scaled_features[b,r,j] = X[b,r,idx[j]] * values[j],  r in [0,C*T)
// idx/values cached in LDS; 64 rows per block; coalesced 2KB row writes.
// ---------------------------------------------------------------------------
__global__ __launch_bounds__(512) void feat_kernel(
    const float* __restrict__ X, const int* __restrict__ idx_ws,
    const float* __restrict__ values_ws, float* __restrict__ out_feat)
{
    __shared__ int   sidx[Kn];
    __shared__ float sval[Kn];
    const int j  = threadIdx.x;
    const int b  = blockIdx.x >> 7;    // 128 row-blocks per batch
    const int rb = blockIdx.x & 127;

    sidx[j] = idx_ws   [b * Kn + j];
    sval[j] = values_ws[b * Kn + j];
    __syncthreads();

    const int   id = sidx[j];
    const float vl = sval[j];
    const float* xb = X        + (size_t)b * Cn * Tn * Vn;
    float*       ob = out_feat + (size_t)b * Cn * Tn * Kn;
    const int r0 = rb * 64;            // C*T / 128 = 64 rows per block
    #pragma unroll 4
    for (int r = r0; r < r0 + 64; ++r) {
        ob[(size_t)r * Kn + j] = xb[(size_t)r * Vn + id] * vl;
    }
}

// ---------------------------------------------------------------------------
// Kernel 4: A_pooled[b,i,j] = A[b, idx[i], idx[j]]
// ---------------------------------------------------------------------------
__global__ __launch_bounds__(512) void adj_kernel(
    const float* __restrict__ A, const int* __restrict__ idx_ws,
    float* __restrict__ out_A)
{
    __shared__ int sidx[Kn];
    const int j  = threadIdx.x;
    const int b  = blockIdx.x >> 5;    // 32 i-blocks per batch, 16 rows each
    const int ib = blockIdx.x & 31;

    sidx[j] = idx_ws[b * Kn + j];
    __syncthreads();

    const float* Ab = A     + (size_t)b * Vn * Vn;
    float*       ob = out_A + (size_t)b * Kn * Kn;
    const int jj = sidx[j];
    for (int i = ib * 16; i < ib * 16 + 16; ++i) {
        ob[(size_t)i * Kn + j] = Ab[(size_t)sidx[i] * Vn + jj];
    }
}

// ---------------------------------------------------------------------------
extern "C" void kernel_launch(void* const* d_in, const int* in_sizes, int n_in,
                              void* d_out, int out_size, void* d_ws, size_t ws_size,
                              hipStream_t stream)
{
    const float* X    = (const float*)d_in[0];   // [B,C,T,V]
    const float* A    = (const float*)d_in[1];   // [B,V,V]
    const float* W    = (const float*)d_in[2];   // [C]
    const float* bias = (const float*)d_in[3];   // scalar

    // Outputs concatenated in return order: A_pooled, scaled_features, idx
    float* out  = (float*)d_out;
    float* outA = out;                                    // B*K*K
    float* outF = outA + (size_t)Bn * Kn * Kn;            // B*C*T*K
    float* outI = outF + (size_t)Bn * Cn * Tn * Kn;       // B*K

    // Workspace: scores [B*V] f32 | values [B*K] f32 | idx [B*K] i32 (~64KB)
    float* scores    = (float*)d_ws;
    float* values_ws = scores + Bn * Vn;
    int*   idx_ws    = (int*)(values_ws + Bn * Kn);

    // 1) scores via f32 WMMA GEMV: 512 waves, 64 blocks of 256 threads
    score_wmma_kernel<<<(Bn * Vn) / 128, 256, 0, stream>>>(X, W, bias, scores);
    // 2) top-K selection, one 1024-thread block per batch
    topk_kernel<<<Bn, 1024, 0, stream>>>(scores, values_ws, idx_ws, outI);
    // 3) feature gather+scale (dominant bandwidth: ~270 MB traffic)
    feat_kernel<<<Bn * 128, 512, 0, stream>>>(X, idx_ws, values_ws, outF);
    // 4) adjacency 2-D gather
    adj_kernel<<<Bn * 32, 512, 0, stream>>>(A, idx_ws, outA);
}